// FFTConv_73555609911734
// MI455X (gfx1250) — compile-verified
//
#include <hip/hip_runtime.h>
#include <math.h>

// ---------------------------------------------------------------------------
// FFT(10000) as four-step (100 x 100) with V_WMMA_F32_16X16X4_F32 GEMMs,
// async global->LDS staging (ASYNCcnt path), then per-segment WMMA GEMM,
// then tiny final contraction.
// ---------------------------------------------------------------------------

typedef __attribute__((ext_vector_type(2))) float v2f;
typedef __attribute__((ext_vector_type(8))) float v8f;
typedef int g_v4i __attribute__((vector_size(16)));  // GCC-style, matches builtin

#define PI_F 3.14159265358979323846f
#define TS   113            // padded table stride (112 logical cols + 1)
#define TROWS 112           // padded rows (100 -> 112, 7 tiles of 16)

// Async global->LDS copy path (CDNA5): guarded so compile never regresses.
#if defined(__gfx1250__) &&                                                   \
    __has_builtin(__builtin_amdgcn_global_load_async_to_lds_b32) &&           \
    __has_builtin(__builtin_amdgcn_global_load_async_to_lds_b128) &&          \
    __has_builtin(__builtin_amdgcn_s_wait_asynccnt)
#define USE_ASYNC_LDS 1
#else
#define USE_ASYNC_LDS 0
#endif

#if USE_ASYNC_LDS
__device__ __forceinline__ void async_cp_b32(const float* g, float* l) {
  __builtin_amdgcn_global_load_async_to_lds_b32(
      (__attribute__((address_space(1))) int*)g,
      (__attribute__((address_space(3))) int*)l, 0, 0);
}
__device__ __forceinline__ void async_cp_b128(const float* g, float* l) {
  __builtin_amdgcn_global_load_async_to_lds_b128(
      (__attribute__((address_space(1))) g_v4i*)g,
      (__attribute__((address_space(3))) g_v4i*)l, 0, 0);
}
__device__ __forceinline__ void async_wait0() {
  __builtin_amdgcn_s_wait_asynccnt(0);
}
#endif

__device__ __forceinline__ v8f wmma4(v2f a, v2f b, v8f c) {
  // 8 args: (neg_a, A, neg_b, B, c_mod, C, reuse_a, reuse_b); f32 NEG must be 0
  return __builtin_amdgcn_wmma_f32_16x16x4_f32(false, a, false, b, (short)0, c,
                                               false, false);
}

// ---------------------------------------------------------------------------
// Kernel 0: DFT-100 cos/sin matrices, zero-padded to 112x112 (stride 113)
// C[i][j] = cos(2*pi*i*j/100), S[i][j] = sin(2*pi*i*j/100)  (symmetric)
// ---------------------------------------------------------------------------
__global__ void init_tables_kernel(float* __restrict__ Cg, float* __restrict__ Sg) {
  int i = blockIdx.x * blockDim.x + threadIdx.x;
  if (i >= TROWS * TS) return;
  int r = i / TS, c = i % TS;
  float cv = 0.0f, sv = 0.0f;
  if (r < 100 && c < 100) {
    int p = (r * c) % 100;  // exact range reduction
    float ang = (2.0f * PI_F / 100.0f) * (float)p;
    __sincosf(ang, &sv, &cv);
  }
  Cg[i] = cv;
  Sg[i] = sv;
}

// ---------------------------------------------------------------------------
// Kernel 1: per-row FFT-10000 + log-magnitude.
//   n = n1 + 100*n2, k = k2 + 100*k1
//   Y[n1,k2]  = DFT100_{n2}(x[n1+100*n2])              (stage A, WMMA)
//   Y'[n1,k2] = Y * W10000^{n1*k2}                      (stage B, VALU)
//   X[k1,k2]  = DFT100_{n1}(Y'[n1,k2])                  (stage C, WMMA)
//   out[k1*100+k2] = 0.5*log(|X|^2)
// One workgroup (256 thr = 8 waves) per (b,c) row; ~240 KB dynamic LDS.
// ---------------------------------------------------------------------------
__global__ __launch_bounds__(256)
void fft_logmag_kernel(const float* __restrict__ x, const float* __restrict__ Cg,
                       const float* __restrict__ Sg, float* __restrict__ lm) {
  extern __shared__ float smem[];
  float* Cl  = smem;                 // TROWS*TS
  float* Sl  = Cl + TROWS * TS;      // TROWS*TS
  float* Xa  = Sl + TROWS * TS;      // 112 x 100 (M-pad x K)
  float* Yre = Xa + TROWS * 100;     // 112 x 112
  float* Yim = Yre + TROWS * TROWS;  // 112 x 112

  const int tid  = threadIdx.x;
  const int lane = tid & 31;
  const int wv   = tid >> 5;
  const int ln15 = lane & 15;
  const int hi   = lane >> 4;
  const int row  = blockIdx.x;  // 0..5119
  const float* xr = x + (size_t)row * 10000;

#if USE_ASYNC_LDS
  // Tables are 16B-aligned and contiguous on both sides -> b128 async DMA.
  for (int i = tid * 4; i < TROWS * TS; i += 1024) {
    async_cp_b128(Cg + i, Cl + i);
    async_cp_b128(Sg + i, Sl + i);
  }
  // Transposed fill: per-lane LDS destination address, b32 async.
  for (int i = tid; i < 10000; i += 256) {
    int n1 = i % 100, n2 = i / 100;
    async_cp_b32(xr + i, &Xa[n1 * 100 + n2]);
  }
  for (int i = 10000 + tid; i < TROWS * 100; i += 256) Xa[i] = 0.0f;  // pad
  async_wait0();
#else
  for (int i = tid; i < TROWS * TS; i += 256) { Cl[i] = Cg[i]; Sl[i] = Sg[i]; }
  for (int i = tid; i < 10000; i += 256) {
    int n1 = i % 100, n2 = i / 100;
    Xa[n1 * 100 + n2] = xr[i];
  }
  for (int i = 10000 + tid; i < TROWS * 100; i += 256) Xa[i] = 0.0f;  // pad
#endif
  __syncthreads();

  // ---- Stage A: Yre = Xa*C, Yim = -(Xa*S). 7x7=49 tiles over 8 waves. ----
  for (int t = wv; t < 49; t += 8) {
    int tm = (t / 7) * 16, tn = (t % 7) * 16;
    v8f ar = {}, ai = {};
#pragma unroll 5
    for (int k0 = 0; k0 < 100; k0 += 4) {
      int ka = k0 + 2 * hi;
      const float* ap = &Xa[(tm + ln15) * 100 + ka];
      v2f a;  a.x = ap[0];  a.y = ap[1];
      const float* cp = &Cl[ka * TS + tn + ln15];
      v2f bc; bc.x = cp[0]; bc.y = cp[TS];
      const float* sp = &Sl[ka * TS + tn + ln15];
      v2f bs; bs.x = sp[0]; bs.y = sp[TS];
      ar = wmma4(a, bc, ar);
      ai = wmma4(a, bs, ai);
    }
    int n = tn + ln15;
#pragma unroll
    for (int j = 0; j < 8; ++j) {
      int m = tm + j + 8 * hi;
      Yre[m * TROWS + n] = ar[j];
      Yim[m * TROWS + n] = -ai[j];
    }
  }
  __syncthreads();

  // ---- Stage B: twiddle by W10000^{n1*k2} = cos(th) - i*sin(th) ----
  for (int i = tid; i < 10000; i += 256) {
    int n1 = i / 100, k2 = i % 100;
    float th = (2.0f * PI_F / 10000.0f) * (float)(n1 * k2);
    float s, c;
    __sincosf(th, &s, &c);
    float re = Yre[n1 * TROWS + k2], im = Yim[n1 * TROWS + k2];
    Yre[n1 * TROWS + k2] = re * c + im * s;
    Yim[n1 * TROWS + k2] = im * c - re * s;
  }
  __syncthreads();

  // ---- Stage C: X[k1][k2] = sum_n1 (cos - i*sin)(Yre + i*Yim); logmag ----
  float* out = lm + (size_t)row * 10000;
  for (int t = wv; t < 49; t += 8) {
    int tm = (t / 7) * 16, tn = (t % 7) * 16;
    v8f xre = {}, xim = {};
#pragma unroll 5
    for (int k0 = 0; k0 < 100; k0 += 4) {
      int ka = k0 + 2 * hi;
      const float* cp = &Cl[(tm + ln15) * TS + ka];
      v2f ac;  ac.x = cp[0];   ac.y = cp[1];
      const float* sp = &Sl[(tm + ln15) * TS + ka];
      v2f as;  as.x = sp[0];   as.y = sp[1];
      v2f asn; asn.x = -as.x;  asn.y = -as.y;
      const float* rp = &Yre[ka * TROWS + tn + ln15];
      v2f br;  br.x = rp[0];   br.y = rp[TROWS];
      const float* ip = &Yim[ka * TROWS + tn + ln15];
      v2f bi;  bi.x = ip[0];   bi.y = ip[TROWS];
      xre = wmma4(ac, br, xre);
      xre = wmma4(as, bi, xre);
      xim = wmma4(ac, bi, xim);
      xim = wmma4(asn, br, xim);
    }
    int k2 = tn + ln15;
    if (k2 < 100) {
#pragma unroll
      for (int j = 0; j < 8; ++j) {
        int k1 = tm + j + 8 * hi;
        if (k1 < 100) {
          float re = xre[j], im = xim[j];
          out[k1 * 100 + k2] = 0.5f * __logf(re * re + im * im);
        }
      }
    }
  }
}

// ---------------------------------------------------------------------------
// Kernel 2: per-segment GEMM. x2[b,o,s] = relu(sum_{c,k} lm[b,c,100s+k]*W1[s,o,c,k] + b1[s,o])
// M=256 (b), N=32 (o), K=2000 in 20 chunks of 100 (one input channel each).
// Grid = 100 segments, 256 threads (8 waves, 4 tiles each of 32 total tiles).
// ---------------------------------------------------------------------------
__global__ __launch_bounds__(256)
void seg_gemm_kernel(const float* __restrict__ lm, const float* __restrict__ W1,
                     const float* __restrict__ b1, float* __restrict__ x2) {
  extern __shared__ float smem[];
  float* As = smem;            // 256 x 100, m-major
  float* Bs = As + 256 * 100;  // 100 x 32, k-major

  const int s = blockIdx.x;
  const int tid = threadIdx.x;
  const int lane = tid & 31, wv = tid >> 5, ln15 = lane & 15, hi = lane >> 4;

  v8f acc[4];
  v8f zero = {};
#pragma unroll
  for (int q = 0; q < 4; ++q) acc[q] = zero;

  for (int c = 0; c < 20; ++c) {
    __syncthreads();
#if USE_ASYNC_LDS
    // A chunk: rows of 100 floats; every 4-float group stays inside a row
    // (10000, 100 and the k step are all multiples of 4) -> b128 async.
    for (int i = tid * 4; i < 256 * 100; i += 1024) {
      int m = i / 100, k = i % 100;
      async_cp_b128(&lm[((size_t)m * 20 + c) * 10000 + s * 100 + k], &As[i]);
    }
    // B chunk is a transposing gather (o fastest in LDS, k fastest in mem).
    for (int i = tid; i < 100 * 32; i += 256) {
      int k = i / 32, o = i % 32;
      async_cp_b32(&W1[(((size_t)s * 32 + o) * 20 + c) * 100 + k], &Bs[i]);
    }
    async_wait0();
#else
    for (int i = tid; i < 256 * 100; i += 256) {
      int m = i / 100, k = i % 100;
      As[i] = lm[((size_t)m * 20 + c) * 10000 + s * 100 + k];
    }
    for (int i = tid; i < 100 * 32; i += 256) {
      int k = i / 32, o = i % 32;
      Bs[i] = W1[(((size_t)s * 32 + o) * 20 + c) * 100 + k];
    }
#endif
    __syncthreads();
    // hint the next K-chunk of the streaming A operand into cache
    if (c + 1 < 20) {
      __builtin_prefetch(&lm[((size_t)tid * 20 + c + 1) * 10000 + s * 100], 0, 1);
    }
#pragma unroll
    for (int q = 0; q < 4; ++q) {
      int t = wv * 4 + q;
      int tm = (t >> 1) * 16;  // b tile
      int tn = (t & 1) * 16;   // o tile
#pragma unroll 5
      for (int k0 = 0; k0 < 100; k0 += 4) {
        int ka = k0 + 2 * hi;
        const float* ap = &As[(tm + ln15) * 100 + ka];
        v2f a; a.x = ap[0]; a.y = ap[1];
        const float* bp = &Bs[ka * 32 + tn + ln15];
        v2f b; b.x = bp[0]; b.y = bp[32];
        acc[q] = wmma4(a, b, acc[q]);
      }
    }
  }
  // epilogue: bias + relu; x2 layout [b][o(32)][s(100)]
#pragma unroll
  for (int q = 0; q < 4; ++q) {
    int t = wv * 4 + q;
    int tm = (t >> 1) * 16, tn = (t & 1) * 16;
    int o = tn + ln15;
    float bias = b1[s * 32 + o];
#pragma unroll
    for (int j = 0; j < 8; ++j) {
      int b_ = tm + j + 8 * hi;
      float v = acc[q][j] + bias;
      x2[(size_t)b_ * 3200 + o * 100 + s] = v > 0.0f ? v : 0.0f;
    }
  }
}

// ---------------------------------------------------------------------------
// Kernel 3: out[b,o] = relu(sum_{j<3200} x2[b,j]*W2[o,j] + b2[o]); 9.8 MFLOP.
// ---------------------------------------------------------------------------
__global__ __launch_bounds__(256)
void final_kernel(const float* __restrict__ x2, const float* __restrict__ W2,
                  const float* __restrict__ b2, float* __restrict__ out) {
  int b_ = threadIdx.x;  // 0..255
  int o  = blockIdx.x;   // 0..5
  const float* xp = x2 + (size_t)b_ * 3200;
  const float* wp = W2 + (size_t)o * 3200;
  float acc = 0.0f;
  for (int j = 0; j < 3200; ++j) acc = fmaf(xp[j], wp[j], acc);
  acc += b2[o];
  out[b_ * 6 + o] = acc > 0.0f ? acc : 0.0f;
}

// ---------------------------------------------------------------------------
extern "C" void kernel_launch(void* const* d_in, const int* in_sizes, int n_in,
                              void* d_out, int out_size, void* d_ws, size_t ws_size,
                              hipStream_t stream) {
  const float* x  = (const float*)d_in[0];  // [256,20,10000]
  const float* W1 = (const float*)d_in[1];  // [100,32,20,100]
  const float* b1 = (const float*)d_in[2];  // [100,32]
  const float* W2 = (const float*)d_in[3];  // [6,32,100]
  const float* b2 = (const float*)d_in[4];  // [6]
  float* out = (float*)d_out;               // [256,6]

  float* ws = (float*)d_ws;
  float* Cg = ws;                             // 112*113
  float* Sg = Cg + TROWS * TS;                // 112*113
  float* lm = Sg + TROWS * TS;                // 5120*10000 (204.8 MB)
  float* x2 = lm + (size_t)5120 * 10000;      // 256*32*100

  // Kernel 0: DFT tables
  int tbl = TROWS * TS;
  init_tables_kernel<<<(tbl + 255) / 256, 256, 0, stream>>>(Cg, Sg);

  // Kernel 1: FFT + log|.| per (b,c) row
  size_t fft_lds = (size_t)(2 * TROWS * TS + TROWS * 100 + 2 * TROWS * TROWS) *
                   sizeof(float);  // ~240.6 KB
  fft_logmag_kernel<<<5120, 256, fft_lds, stream>>>(x, Cg, Sg, lm);

  // Kernel 2: per-segment WMMA GEMM + bias + relu
  size_t gemm_lds = (size_t)(256 * 100 + 100 * 32) * sizeof(float);  // 115.2 KB
  seg_gemm_kernel<<<100, 256, gemm_lds, stream>>>(lm, W1, b1, x2);

  // Kernel 3: final contraction + relu
  final_kernel<<<6, 256, 0, stream>>>(x2, W2, b2, out);
}